// Sinkhorn_28028956573873
// MI455X (gfx1250) — compile-verified
//
#include <hip/hip_runtime.h>

typedef float v8f __attribute__((ext_vector_type(8)));
typedef float v4f __attribute__((ext_vector_type(4)));
typedef float v2f __attribute__((ext_vector_type(2)));

namespace {

constexpr int   kN   = 1024;
constexpr int   kM   = 1024;
constexpr float kEps = 1e-4f;

__device__ __forceinline__ float wave_reduce_add(float v) {
#pragma unroll
  for (int off = 16; off > 0; off >>= 1) v += __shfl_xor(v, off, 32);
  return v;
}

// Sum of all 1024 LDS floats, computed redundantly per wave (cheap, no barrier).
__device__ __forceinline__ float lds_sum_all(const float* a, int lane) {
  float v = 0.0f;
#pragma unroll
  for (int i = 0; i < kN / 32; ++i) v += a[lane + 32 * i];
  return wave_reduce_add(v);
}

// ---------------------------------------------------------------------------
// WMMA column-direction reduction core (shared by both code paths):
//   U[m] = sum_{n<rows4} R[n]*s[n][m] ;  returns this lane's U[m0+lane]
// A (16x4 f32) = R[n0..n0+3] broadcast to all 16 rows; B (4x16 f32) = row-major
// s tile. All D rows are identical, so VGPR0 of the accumulator holds U.
// ---------------------------------------------------------------------------
template <bool USER>
__device__ __forceinline__ float wmma_col_sum(const float* __restrict__ S,
                                              const float* __restrict__ ldsR,
                                              int rows4, int m0, int lane) {
  v8f acc0 = {};
  v8f acc1 = {};
  const int rsel = (lane >> 4) << 1;  // 0 for lanes 0-15, 2 for lanes 16-31
  const float* p0 = S + (size_t)rsel * kM + m0 + (lane & 15);
#pragma unroll 2
  for (int n0 = 0; n0 < rows4; n0 += 4) {
    v2f a;
    if (USER) {
      a = *(const v2f*)&ldsR[n0 + rsel];  // ds_load_b64: R[n0+rsel], R[n0+rsel+1]
    } else {
      a.x = 1.0f;  // iteration 0: unweighted column sums over ALL rows
      a.y = 1.0f;
    }
    const float* p = p0 + (size_t)n0 * kM;
    v2f b0, b1;
    b0.x = p[0];        // rows n0+rsel,   cols m0..m0+15
    b0.y = p[kM];       // rows n0+rsel+1
    b1.x = p[16];       // cols m0+16..m0+31
    b1.y = p[kM + 16];
    acc0 = __builtin_amdgcn_wmma_f32_16x16x4_f32(false, a, false, b0, (short)0,
                                                 acc0, false, false);
    acc1 = __builtin_amdgcn_wmma_f32_16x16x4_f32(false, a, false, b1, (short)0,
                                                 acc1, false, false);
  }
  // acc0[0]: lanes 0-15 -> U[m0+lane]; acc1[0]: lanes 16-31 -> U[m0+lane]
  return (lane < 16) ? acc0[0] : acc1[0];
}

// ===========================================================================
// Multi-kernel path: one launch per Sinkhorn pass (kernel boundary = global
// barrier), R/C factor vectors live in d_ws. Small workgroups spread over the
// whole device so every pass runs at full-machine bandwidth.
// ===========================================================================

// C[b][m] = (m<nc) ? 1 / (sum_{n<rows4} R[n]*s[n][m] + eps*sumR) : 0
template <bool USER>
__launch_bounds__(256)
__global__ void col_pass_kernel(const float* __restrict__ s,
                                const float* __restrict__ Rws,
                                float* __restrict__ Cws,
                                const int* __restrict__ nrows,
                                const int* __restrict__ ncols) {
  __shared__ __align__(16) float ldsR[kN];
  const int b    = blockIdx.y;
  const int tid  = threadIdx.x;
  const int lane = tid & 31;
  const int wave = tid >> 5;
  const int nr   = nrows[b];
  const int nc   = ncols[b];
  const float* S = s + (size_t)b * kN * kM;
  float*       C = Cws + (size_t)b * kM;

  float sumR;
  int   rows4;
  if (USER) {
    const float* R = Rws + (size_t)b * kN;
#pragma unroll
    for (int i = 0; i < kN / 256; ++i) ldsR[tid + 256 * i] = R[tid + 256 * i];
    __syncthreads();
    sumR  = lds_sum_all(ldsR, lane);
    rows4 = (nr + 3) & ~3;  // R[n]=0 for n>=nr makes the 4-padding exact
  } else {
    sumR  = (float)kN;      // reference sums over ALL rows on iteration 0
    rows4 = kN;
  }

  const int m0 = (blockIdx.x * 8 + wave) * 32;  // this wave owns 32 columns
  const int m  = m0 + lane;
  float cv = 0.0f;
  if (m0 < nc) {  // tiles fully past ncols: skip all reads
    const float u = wmma_col_sum<USER>(S, ldsR, rows4, m0, lane) + kEps * sumR;
    if (m < nc) cv = 1.0f / u;
  }
  C[m] = cv;
}

// R[b][n] = (n<nr) ? 1 / (sum_{m<nc} s[n][m]*C[m] + eps*sumC) : 0
__launch_bounds__(256)
__global__ void row_pass_kernel(const float* __restrict__ s,
                                const float* __restrict__ Cws,
                                float* __restrict__ Rws,
                                const int* __restrict__ nrows,
                                const int* __restrict__ ncols) {
  __shared__ __align__(16) float ldsC[kM];
  const int b    = blockIdx.y;
  const int tid  = threadIdx.x;
  const int lane = tid & 31;
  const int wave = tid >> 5;
  const int nr   = nrows[b];
  const int nc   = ncols[b];
  const float* S = s + (size_t)b * kN * kM;
  const float* C = Cws + (size_t)b * kM;
  float*       R = Rws + (size_t)b * kN;

#pragma unroll
  for (int i = 0; i < kM / 256; ++i) ldsC[tid + 256 * i] = C[tid + 256 * i];
  __syncthreads();
  const float sumC = lds_sum_all(ldsC, lane);

  const int gw   = blockIdx.x * 8 + wave;  // 64 waves cover the 1024 rows
  const int mpad = (nc + 127) & ~127;      // C[m]=0 for m>=nc => padding exact
  for (int row = gw; row < kN; row += 64) {
    if (row < nr) {
      const float* p = S + (size_t)row * kM;
      v4f acc = {};
      for (int mm = lane * 4; mm < mpad; mm += 128) {
        const v4f sv = *(const v4f*)(p + mm);          // global_load_b128
        const v4f cvv = *(const v4f*)(&ldsC[mm]);      // ds_load_b128
        acc += sv * cvv;
      }
      float t = wave_reduce_add((acc.x + acc.y) + (acc.z + acc.w));
      t += kEps * sumC;
      if (lane == 0) R[row] = 1.0f / t;
    } else {
      if (lane == 0) R[row] = 0.0f;  // zero masked tail rows
    }
  }
}

// out = mask ? (s+eps)*R[n]*C[m] : 0    (b128 nontemporal streaming)
__launch_bounds__(256)
__global__ void final_pass_kernel(const float* __restrict__ s,
                                  const float* __restrict__ Rws,
                                  const float* __restrict__ Cws,
                                  const int* __restrict__ nrows,
                                  const int* __restrict__ ncols,
                                  float* __restrict__ out) {
  __shared__ float ldsR[64];
  const int b    = blockIdx.y;
  const int tid  = threadIdx.x;
  const int nr   = nrows[b];
  const int nc   = ncols[b];
  const float* S = s + (size_t)b * kN * kM;
  float*       O = out + (size_t)b * kN * kM;
  const float* R = Rws + (size_t)b * kN;
  const float* C = Cws + (size_t)b * kM;

  const int row0 = blockIdx.x * 64;  // this workgroup covers 64 rows
  if (tid < 64) ldsR[tid] = R[row0 + tid];
  const int  c4   = tid * 4;         // 256 threads x 4 cols = full row
  const bool live = c4 < nc;         // C components past nc are already 0
  v4f cf = {};
  if (live) cf = *(const v4f*)(C + c4);
  __syncthreads();

  for (int r = 0; r < 64; ++r) {
    const int row = row0 + r;
    v4f v = {};
    if (live && row < nr) {
      const v4f sv = __builtin_nontemporal_load((const v4f*)(S + (size_t)row * kM + c4));
      v = (sv + kEps) * ldsR[r] * cf;
    }
    __builtin_nontemporal_store(v, (v4f*)(O + (size_t)row * kM + c4));
  }
}

// ===========================================================================
// Fallback (ws too small): persistent one-workgroup-per-batch kernel.
// ===========================================================================

__device__ __forceinline__ void p_col_pass(const float* __restrict__ S,
                                           const float* __restrict__ ldsR,
                                           float* __restrict__ ldsC, int rows4,
                                           float epsTerm, int nc, int lane,
                                           int wave, bool useR) {
  const int m0 = wave * 32;
  const int m  = m0 + lane;
  float cv = 0.0f;
  if (m0 < nc) {
    float u = useR ? wmma_col_sum<true>(S, ldsR, rows4, m0, lane)
                   : wmma_col_sum<false>(S, ldsR, rows4, m0, lane);
    u += epsTerm;
    if (m < nc) cv = 1.0f / u;
  }
  ldsC[m] = cv;
}

__device__ __forceinline__ void p_row_pass(const float* __restrict__ S,
                                           float* __restrict__ ldsR,
                                           const float* __restrict__ ldsC,
                                           int nr, int nc, float sumC, int lane,
                                           int wave) {
  const int mpad = (nc + 127) & ~127;
  for (int row = wave; row < nr; row += 32) {
    const float* p = S + (size_t)row * kM;
    v4f acc = {};
    for (int mm = lane * 4; mm < mpad; mm += 128) {
      acc += (*(const v4f*)(p + mm)) * (*(const v4f*)(&ldsC[mm]));
    }
    float t = wave_reduce_add((acc.x + acc.y) + (acc.z + acc.w)) + kEps * sumC;
    if (lane == 0) ldsR[row] = 1.0f / t;
  }
}

__launch_bounds__(1024, 1)
__global__ void sinkhorn_persistent_kernel(const float* __restrict__ s,
                                           const int* __restrict__ nrows,
                                           const int* __restrict__ ncols,
                                           float* __restrict__ out) {
  __shared__ __align__(16) float ldsR[kN];
  __shared__ __align__(16) float ldsC[kM];

  const int b    = blockIdx.x;
  const int tid  = threadIdx.x;
  const int lane = tid & 31;
  const int wave = tid >> 5;
  const int nr   = nrows[b];
  const int nc   = ncols[b];
  const float* S = s + (size_t)b * kN * kM;
  float*       O = out + (size_t)b * kN * kM;

  p_col_pass(S, ldsR, ldsC, kN, kEps * (float)kN, nc, lane, wave, false);
  __syncthreads();
  const int rows4 = (nr + 3) & ~3;
  for (int it = 0; it < 4; ++it) {
    const float sumC = lds_sum_all(ldsC, lane);
    p_row_pass(S, ldsR, ldsC, nr, nc, sumC, lane, wave);
    if (tid >= nr) ldsR[tid] = 0.0f;
    __syncthreads();
    const float sumR = lds_sum_all(ldsR, lane);
    p_col_pass(S, ldsR, ldsC, rows4, kEps * sumR, nc, lane, wave, true);
    __syncthreads();
  }
  {
    const float sumC = lds_sum_all(ldsC, lane);
    p_row_pass(S, ldsR, ldsC, nr, nc, sumC, lane, wave);
    if (tid >= nr) ldsR[tid] = 0.0f;
    __syncthreads();
  }
  const float cfac = ldsC[tid];
  const bool  cin  = tid < nc;
  for (int row = 0; row < nr; ++row) {
    float v = 0.0f;
    if (cin) {
      v = (__builtin_nontemporal_load(&S[(size_t)row * kM + tid]) + kEps) *
          ldsR[row] * cfac;
    }
    __builtin_nontemporal_store(v, &O[(size_t)row * kM + tid]);
  }
  for (int row = nr; row < kN; ++row) {
    __builtin_nontemporal_store(0.0f, &O[(size_t)row * kM + tid]);
  }
}

}  // namespace

extern "C" void kernel_launch(void* const* d_in, const int* in_sizes, int n_in,
                              void* d_out, int out_size, void* d_ws,
                              size_t ws_size, hipStream_t stream) {
  const float* s     = (const float*)d_in[0];
  const int*   nrows = (const int*)d_in[1];
  const int*   ncols = (const int*)d_in[2];
  float*       out   = (float*)d_out;
  const int B = in_sizes[1];
  if (B <= 0) return;

  const size_t ws_need = (size_t)B * (kN + kM) * sizeof(float);
  if (d_ws != nullptr && ws_size >= ws_need) {
    float* Rws = (float*)d_ws;                   // [B][kN]
    float* Cws = (float*)d_ws + (size_t)B * kN;  // [B][kM]
    const dim3 blk(256);
    const dim3 gCol(kM / 256, B);   // 8 waves x 32 cols = 256 cols per WG
    const dim3 gRow(8, B);          // 64 waves cover 1024 rows
    const dim3 gFin(kN / 64, B);    // 64 rows per WG

    // i = 0: column normalization, sums over ALL rows (reference semantics)
    col_pass_kernel<false><<<gCol, blk, 0, stream>>>(s, Rws, Cws, nrows, ncols);
    // i = 1..8: (row, col) x 4
    for (int it = 0; it < 4; ++it) {
      row_pass_kernel<<<gRow, blk, 0, stream>>>(s, Cws, Rws, nrows, ncols);
      col_pass_kernel<true><<<gCol, blk, 0, stream>>>(s, Rws, Cws, nrows, ncols);
    }
    // i = 9: final row normalization, then materialize
    row_pass_kernel<<<gRow, blk, 0, stream>>>(s, Cws, Rws, nrows, ncols);
    final_pass_kernel<<<gFin, blk, 0, stream>>>(s, Rws, Cws, nrows, ncols, out);
  } else {
    sinkhorn_persistent_kernel<<<B, 1024, 0, stream>>>(s, nrows, ncols, out);
  }
}